// CustomSNNLoss_36429912604816
// MI455X (gfx1250) — compile-verified
//
#include <hip/hip_runtime.h>
#include <hip/hip_bf16.h>

// ---------------- problem constants (match reference) ----------------
#define N_PTS       6144
#define DIMS        128
#define NUM_CLASSES 20
#define NUM_BATCHES 5
#define MIN_T       0.1f
#define MAX_T       1.0f
#define INV_TEMP_B  2.0f        // 1 / TEMP_BATCH(0.5)
#define EPS_NORM    1e-8f

// ---------------- tiling ----------------
#define ROWS_PER_BLOCK 128                  // 8 waves * 16 rows
#define COL_TILE       16
#define NUM_COL_TILES  (N_PTS / COL_TILE)   // 384
#define COL_CHUNKS     8
#define TILES_PER_CHUNK (NUM_COL_TILES / COL_CHUNKS)  // 48
#define LDS_STRIDE     132                  // 128 + 4 pad floats (16B aligned, bank-spread)

#define XN_FLOATS (N_PTS * DIMS)

typedef __attribute__((ext_vector_type(2))) float v2f;
typedef __attribute__((ext_vector_type(8))) float v8f;

__device__ __forceinline__ float reduce16(float v) {
  // reduce across the 16 lanes sharing one output row (xor stays inside group)
  v += __shfl_xor(v, 1, 32);
  v += __shfl_xor(v, 2, 32);
  v += __shfl_xor(v, 4, 32);
  v += __shfl_xor(v, 8, 32);
  return v;
}

// ---------------- kernel 1: row-normalize x -> xn (one wave per row) ----------
__global__ void snn_prep(const float* __restrict__ x, float* __restrict__ xn) {
  const int wave = threadIdx.x >> 5;
  const int lane = threadIdx.x & 31;
  const int row  = blockIdx.x * 8 + wave;
  const float4 a = ((const float4*)(x + row * DIMS))[lane];  // 32 lanes * 4 = 128
  float ss = a.x * a.x + a.y * a.y + a.z * a.z + a.w * a.w;
  ss += __shfl_xor(ss, 1, 32);
  ss += __shfl_xor(ss, 2, 32);
  ss += __shfl_xor(ss, 4, 32);
  ss += __shfl_xor(ss, 8, 32);
  ss += __shfl_xor(ss, 16, 32);
  const float inv = 1.0f / fmaxf(sqrtf(ss), EPS_NORM);
  float4 o;
  o.x = a.x * inv; o.y = a.y * inv; o.z = a.z * inv; o.w = a.w * inv;
  ((float4*)(xn + row * DIMS))[lane] = o;
}

// ---------------- kernel 2: zero row accumulators ----------------
__global__ void snn_zero(float* __restrict__ acc) {
  const int i = blockIdx.x * blockDim.x + threadIdx.x;
  if (i < 6 * N_PTS) acc[i] = 0.0f;
}

// ---------------- kernel 3: fused sim-tile WMMA + exp + masked row sums ------
__global__ void __launch_bounds__(256)
snn_main(const float* __restrict__ xn, const int* __restrict__ targets,
         const int* __restrict__ batch0, const float* __restrict__ temp,
         float* __restrict__ rowacc) {
  __shared__ float ldsB[COL_TILE * LDS_STRIDE];

  const int tid  = threadIdx.x;
  const int lane = tid & 31;
  const int wave = tid >> 5;
  const int hi   = lane >> 4;   // 0/1 half-wave
  const int lo   = lane & 15;

  const float t     = fminf(fmaxf(temp[0], MIN_T), MAX_T);
  const float inv_t = 1.0f / t;

  const int rowBase = blockIdx.x * ROWS_PER_BLOCK + wave * 16;

  // A fragment (16x4 fp32 per step): lane holds row rowBase+lo, K pair {4k+2*hi, +1}
  const int    arow = rowBase + lo;
  const float* arp  = xn + arow * DIMS + 2 * hi;
  v2f afrag[32];
#pragma unroll
  for (int kk = 0; kk < 32; ++kk)
    afrag[kk] = *(const v2f*)(arp + kk * 4);

  // row labels for the 8 C-fragment row slots of this lane
  int rowi[8], trow[8], brow[8];
#pragma unroll
  for (int v = 0; v < 8; ++v) {
    rowi[v] = rowBase + v + 8 * hi;
    trow[v] = targets[rowi[v]];
    brow[v] = batch0[rowi[v]];
  }

  float aPT[8], aNT[8], aPB[8], aNB[8], aCP[8], aCB[8];
#pragma unroll
  for (int v = 0; v < 8; ++v) {
    aPT[v] = aNT[v] = aPB[v] = aNB[v] = aCP[v] = aCB[v] = 0.0f;
  }

  const int ctBeg = blockIdx.y * TILES_PER_CHUNK;
  const int ctEnd = ctBeg + TILES_PER_CHUNK;

  const int scol = tid >> 4;   // staging: column 0..15
  const int sseg = tid & 15;   // staging: 8-float segment within column

  for (int ct = ctBeg; ct < ctEnd; ++ct) {
    const int colBase = ct * COL_TILE;

    // stage 16 column vectors (16 x 128 fp32 = 8 KB) into LDS
    {
      const float4* g = (const float4*)(xn + (colBase + scol) * DIMS + sseg * 8);
      const float4 p0 = g[0];
      const float4 p1 = g[1];
      float4* l = (float4*)(ldsB + scol * LDS_STRIDE + sseg * 8);
      l[0] = p0;
      l[1] = p1;
      if (ct + 1 < ctEnd)  // gfx1250 global_prefetch for the next tile
        __builtin_prefetch(xn + (colBase + COL_TILE + scol) * DIMS + sseg * 8, 0, 0);
    }
    __syncthreads();

    // 16x16 sim tile, full fp32: 32 x v_wmma_f32_16x16x4_f32 over K=128
    v8f c = {};
#pragma unroll
    for (int kk = 0; kk < 32; ++kk) {
      const v2f b = *(const v2f*)(ldsB + lo * LDS_STRIDE + kk * 4 + 2 * hi);
      c = __builtin_amdgcn_wmma_f32_16x16x4_f32(false, afrag[kk], false, b,
                                                (short)0, c, false, false);
    }

    // fused postprocess: exp + masks + per-row partial sums
    const int colIdx = colBase + lo;
    const int tcol   = targets[colIdx];
    const int bcol   = batch0[colIdx];
#pragma unroll
    for (int v = 0; v < 8; ++v) {
      const float s  = c[v];
      const float et = __expf(s * inv_t);
      const float eb = __expf(s * INV_TEMP_B);
      const bool st = (trow[v] == tcol);
      const bool sb = (brow[v] == bcol);
      const bool ns = (rowi[v] != colIdx);          // exclude diagonal from pos sums
      aPT[v] += (st && ns)       ? et : 0.0f;       // target positive (no self)
      aNT[v] += st               ? 0.0f : et;       // target negative
      aPB[v] += (st && sb && ns) ? eb : 0.0f;       // batch positive (no self)
      aNB[v] += (st && !sb)      ? eb : 0.0f;       // batch negative
      aCP[v] += st               ? 1.0f : 0.0f;     // cnt same_t (incl self)
      aCB[v] += (st && sb)       ? 1.0f : 0.0f;     // cnt same_t & same_b (incl self)
    }
    __syncthreads();
  }

  // reduce across the 16 lanes of each row, accumulate into global per-row sums
#pragma unroll
  for (int v = 0; v < 8; ++v) {
    const float pt = reduce16(aPT[v]);
    const float nt = reduce16(aNT[v]);
    const float pb = reduce16(aPB[v]);
    const float nb = reduce16(aNB[v]);
    const float cp = reduce16(aCP[v]);
    const float cb = reduce16(aCB[v]);
    if (lo == 0) {
      const int r = rowi[v];
      atomicAdd(&rowacc[0 * N_PTS + r], pt);
      atomicAdd(&rowacc[1 * N_PTS + r], nt);
      atomicAdd(&rowacc[2 * N_PTS + r], pb);
      atomicAdd(&rowacc[3 * N_PTS + r], nb);
      atomicAdd(&rowacc[4 * N_PTS + r], cp);
      atomicAdd(&rowacc[5 * N_PTS + r], cb);
    }
  }
}

// ---------------- kernel 4: per-row losses + weighted class means -> scalar --
__global__ void snn_finalize(const float* __restrict__ rowacc,
                             const int* __restrict__ targets,
                             const int* __restrict__ batch0,
                             const float* __restrict__ wt,
                             const float* __restrict__ wb,
                             float* __restrict__ out) {
  __shared__ float st_sum[NUM_CLASSES], st_cnt[NUM_CLASSES];
  __shared__ float sb_sum[NUM_BATCHES], sb_cnt[NUM_BATCHES];
  const int tid = threadIdx.x;
  if (tid < NUM_CLASSES) { st_sum[tid] = 0.0f; st_cnt[tid] = 0.0f; }
  if (tid < NUM_BATCHES) { sb_sum[tid] = 0.0f; sb_cnt[tid] = 0.0f; }
  __syncthreads();

  for (int i = tid; i < N_PTS; i += blockDim.x) {
    const float pt   = rowacc[0 * N_PTS + i];
    const float nt   = rowacc[1 * N_PTS + i];
    const float pb   = rowacc[2 * N_PTS + i];
    const float nb   = rowacc[3 * N_PTS + i];
    const float cp   = rowacc[4 * N_PTS + i];   // includes self
    const float cb   = rowacc[5 * N_PTS + i];   // includes self
    const float cn   = (float)N_PTS - cp;

    if (cp >= 2.0f && cn >= 1.0f) {             // valid target row
      const float li = -__logf(pt / (pt + nt));
      atomicAdd(&st_sum[targets[i]], li);
      atomicAdd(&st_cnt[targets[i]], 1.0f);
    }
    const float cnb = cp - cb;                  // same_t & !same_b count
    if (cb >= 2.0f && cnb >= 1.0f) {            // valid batch row
      const float lbi = -__logf(pb / (pb + nb));
      atomicAdd(&sb_sum[batch0[i]], 1.0f / lbi);
      atomicAdd(&sb_cnt[batch0[i]], 1.0f);
    }
  }
  __syncthreads();

  if (tid == 0) {
    float lt = 0.0f, lb = 0.0f;
    for (int c = 0; c < NUM_CLASSES; ++c)
      if (st_cnt[c] > 0.0f) lt += (st_sum[c] / fmaxf(st_cnt[c], 1.0f)) * wt[c];
    for (int c = 0; c < NUM_BATCHES; ++c)
      if (sb_cnt[c] > 0.0f) lb += (sb_sum[c] / fmaxf(sb_cnt[c], 1.0f)) * wb[c];
    out[0] = 0.9f * lt + 0.1f * lb;
  }
}

// ---------------- launcher ----------------
extern "C" void kernel_launch(void* const* d_in, const int* in_sizes, int n_in,
                              void* d_out, int out_size, void* d_ws, size_t ws_size,
                              hipStream_t stream) {
  const float* x       = (const float*)d_in[0];
  const float* temp    = (const float*)d_in[1];
  const float* wt      = (const float*)d_in[2];
  const float* wb      = (const float*)d_in[3];
  const int*   targets = (const int*)d_in[4];
  const int*   batch0  = (const int*)d_in[5];

  float* xn     = (float*)d_ws;
  float* rowacc = xn + XN_FLOATS;

  snn_prep<<<N_PTS / 8, 256, 0, stream>>>(x, xn);
  snn_zero<<<(6 * N_PTS + 255) / 256, 256, 0, stream>>>(rowacc);

  dim3 grid(N_PTS / ROWS_PER_BLOCK, COL_CHUNKS);   // 48 x 8 = 384 workgroups
  snn_main<<<grid, 256, 0, stream>>>(xn, targets, batch0, temp, rowacc);

  snn_finalize<<<1, 1024, 0, stream>>>(rowacc, targets, batch0, wt, wb,
                                       (float*)d_out);
}